// ContextualViewModel_30880814858753
// MI455X (gfx1250) — compile-verified
//
#include <hip/hip_runtime.h>

// ext-vector types for WMMA operands (f32 16x16x4: A=16x4 -> 2 VGPRs/lane,
// B=4x16 -> 2 VGPRs/lane, C/D=16x16 f32 -> 8 VGPRs/lane, wave32).
typedef __attribute__((ext_vector_type(2))) float v2f;
typedef __attribute__((ext_vector_type(4))) float v4f;
typedef __attribute__((ext_vector_type(8))) float v8f;

#define HH 512
#define WW 512
#define KK 128   // context length == channel count C

// ---------------------------------------------------------------------------
// Kernel 1 (single block, 256 threads = 8 waves):
//   Stage 1: g[c] = sum_k x[sx_k, sy_k, c]          (gather-sum into LDS)
//   Stage 2: s = g @ W via V_WMMA_F32_16X16X4_F32   (exact fp32 path)
// Wave w computes s[16w .. 16w+15]; A is g replicated across the 16 M rows,
// so every row of D equals s; we read row M=0 (acc VGPR0, lanes 0..15).
// ---------------------------------------------------------------------------
__global__ void __launch_bounds__(256)
cvm_reduce_vecmat_kernel(const float* __restrict__ x,
                         const float* __restrict__ Wm,
                         const int*   __restrict__ sid,
                         float*       __restrict__ s_out) {
    __shared__ int   sid_lds[KK];
    __shared__ float part[256];
    __shared__ float gsum[KK];

    const int tid = threadIdx.x;
    if (tid < KK) sid_lds[tid] = sid[tid];
    __syncthreads();

    // ---- Stage 1: gather-sum. thread t handles channel c = t & 127 over
    // half the k-range (t<128 -> k 0..63, t>=128 -> k 64..127). Loads are
    // fully coalesced (128 consecutive floats per gathered row).
    const int c  = tid & (KK - 1);
    const int k0 = (tid >> 7) * (KK / 2);
    float acc = 0.0f;
#pragma unroll 8
    for (int k = 0; k < KK / 2; ++k) {
        const int id = sid_lds[k0 + k];
        const size_t base = (size_t)(id >> 9) * (size_t)(WW * KK)   // sx = id / 512
                          + (size_t)(id & (WW - 1)) * (size_t)KK;   // sy = id % 512
        acc += x[base + c];
    }
    part[tid] = acc;
    __syncthreads();
    if (tid < KK) gsum[tid] = part[tid] + part[tid + KK];
    __syncthreads();

    // ---- Stage 2: s = g (1x128) @ W (128x128), tiled 16(N) x 4(K) per wave.
    // A 16x4 layout: lanes 0..15 -> {K=0,K=1}, lanes 16..31 -> {K=2,K=3},
    // M = lane%16 (we replicate g over M). B 4x16: lanes = N, same K split.
    const int wave = tid >> 5;            // 0..7 -> N tile
    const int lane = tid & 31;
    const int n0   = wave * 16;
    const int nn   = lane & 15;
    const int kofs = (lane < 16) ? 0 : 2;

    v8f cacc = {};
    for (int kk = 0; kk < KK; kk += 4) {
        v2f a, b;
        a.x = gsum[kk + kofs];
        a.y = gsum[kk + kofs + 1];
        b.x = Wm[(size_t)(kk + kofs)     * KK + n0 + nn];
        b.y = Wm[(size_t)(kk + kofs + 1) * KK + n0 + nn];
        // (neg_a, A, neg_b, B, c_mod, C, reuse_a, reuse_b)
        cacc = __builtin_amdgcn_wmma_f32_16x16x4_f32(
            false, a, false, b, (short)0, cacc, false, false);
    }
    // Row M=0 of D: accumulator VGPR0, lanes 0..15.
    if (lane < 16) s_out[n0 + nn] = cacc[0];
}

// ---------------------------------------------------------------------------
// Kernel 2: broadcast s (128 floats) to all 512*512 positions.
// Pure 128-bit stores; grid stride is a multiple of 32 float4s so each
// thread's 16-byte slice of s is loop-invariant. This is the HBM-bound part
// (134 MB out -> ~5.8 us floor at 23.3 TB/s).
// ---------------------------------------------------------------------------
__global__ void __launch_bounds__(256)
cvm_broadcast_kernel(const v4f* __restrict__ s4,
                     v4f*       __restrict__ out,
                     unsigned n4) {
    const unsigned tid    = blockIdx.x * blockDim.x + threadIdx.x;
    const unsigned stride = gridDim.x * blockDim.x;   // multiple of 32
    const v4f v = s4[tid & 31u];
    for (unsigned i = tid; i < n4; i += stride) {
        out[i] = v;
    }
}

extern "C" void kernel_launch(void* const* d_in, const int* in_sizes, int n_in,
                              void* d_out, int out_size, void* d_ws, size_t ws_size,
                              hipStream_t stream) {
    (void)in_sizes; (void)n_in; (void)ws_size;

    const float* x   = (const float*)d_in[0];   // (512, 512, 128) fp32
    const float* Wm  = (const float*)d_in[1];   // (128, 128) fp32
    const int*   sid = (const int*)d_in[2];     // (128,) int

    float* s = (float*)d_ws;                    // 128-float scratch vector

    cvm_reduce_vecmat_kernel<<<1, 256, 0, stream>>>(x, Wm, sid, s);

    const unsigned n4 = (unsigned)(out_size / 4);   // 8,388,608 float4s
    cvm_broadcast_kernel<<<4096, 256, 0, stream>>>(
        (const v4f*)s, (v4f*)d_out, n4);
}